// BackgroundNoiseLayer_20830591386289
// MI455X (gfx1250) — compile-verified
//
#include <hip/hip_runtime.h>

// ---- problem constants (from reference) ------------------------------------
#define N_V1   200000
#define N_LM   50000
#define N_TOT  (N_V1 + N_LM)      // 250000
#define T_LEN  600
#define T_PAD  608                // 38 * 16
#define U_RAW  10
#define KP     16                 // K padded to 16 (zeros) -> 4 WMMA k-steps
#define NNZ_V1 800000
#define NNZ_LM 200000

typedef __attribute__((ext_vector_type(2))) float v2f;
typedef __attribute__((ext_vector_type(8))) float v8f;

// ---- kernel 1: zero padded dense W -----------------------------------------
__global__ void zero_w_kernel(float4* __restrict__ w, int n4) {
    int i = blockIdx.x * blockDim.x + threadIdx.x;
    if (i < n4) w[i] = make_float4(0.f, 0.f, 0.f, 0.f);
}

// ---- kernel 2: COO scatter with duplicate-sum (tf.sparse semantics) --------
__global__ void scatter_w_kernel(const float* __restrict__ wv1,
                                 const int* __restrict__ rv1,
                                 const int* __restrict__ cv1,
                                 const float* __restrict__ wlm,
                                 const int* __restrict__ rlm,
                                 const int* __restrict__ clm,
                                 float* __restrict__ W) {
    int i = blockIdx.x * blockDim.x + threadIdx.x;
    if (i < NNZ_V1) {
        atomicAdd(&W[(size_t)rv1[i] * KP + cv1[i]], wv1[i]);
    } else if (i < NNZ_V1 + NNZ_LM) {
        int j = i - NNZ_V1;
        atomicAdd(&W[((size_t)N_V1 + rlm[j]) * KP + clm[j]], wlm[j]);
    }
}

// ---- kernel 3: deterministic Poisson(1) spikes, padded to (T_PAD, KP) ------
__device__ __forceinline__ unsigned mix32(unsigned x) {
    x ^= x >> 16; x *= 0x7feb352dU;
    x ^= x >> 15; x *= 0x846ca68bU;
    x ^= x >> 16;
    return x;
}

__global__ void gen_spikes_kernel(float* __restrict__ S) {
    int i = blockIdx.x * blockDim.x + threadIdx.x;   // i = t*KP + u
    if (i >= T_PAD * KP) return;
    int t = i / KP;
    int u = i - t * KP;
    float v = 0.f;
    if (t < T_LEN && u < U_RAW) {
        // Knuth Poisson sampler, lambda = 1 (deterministic hash-seeded xorshift)
        unsigned st = mix32((unsigned)(t * U_RAW + u) * 0x9E3779B9u + 42u);
        if (st == 0u) st = 0x1234567u;
        const float L = 0.36787944117144233f;        // exp(-1)
        float p = 1.0f;
        int k = 0;
        do {
            st ^= st << 13; st ^= st >> 17; st ^= st << 5;
            float uu = (float)(st >> 8) * (1.0f / 16777216.0f);
            p *= uu;
            ++k;
        } while (p > L && k < 32);
        v = (float)(k - 1);
    }
    S[i] = v;
}

// ---- kernel 4: out(T_PAD x N_TOT) = S(T_PAD x 16) * W^T(16 x N_TOT) --------
// One wave per 16x16 output tile; 4x V_WMMA_F32_16X16X4_F32 (full fp32 path).
// 160-thread blocks = 5 waves over 80 consecutive neurons.
// Stores: per-D-VGPR the wave writes two 64B-aligned, 64B-contiguous segments
// (full cachelines). Non-temporal hint keeps the 16MB W resident in L2 while
// the 600MB output streams through.
__global__ __launch_bounds__(160)
void bkg_gemm_kernel(const float* __restrict__ S,
                     const float* __restrict__ W,
                     float* __restrict__ out) {
    const int lane  = threadIdx.x & 31;
    const int wave  = threadIdx.x >> 5;
    const int nbase = (blockIdx.x * 5 + wave) * 16;   // neuron tile base
    const int mbase = blockIdx.y * 16;                // time tile base
    const int r = lane & 15;       // row within half
    const int h = lane >> 4;       // half: selects K parity pair

    // ISA A(16x4 f32) layout: lane holds {A[m][k0], A[m][k0+1]}, k0 = kc + 2h.
    // B(4x16) mirrors it: lane holds {B[k0][n], B[k0+1][n]} = W[n][k0..k0+1].
    const float* arow = S + (size_t)(mbase + r) * KP + 2 * h;
    const float* brow = W + (size_t)(nbase + r) * KP + 2 * h;

    v8f acc = {};
#pragma unroll
    for (int kc = 0; kc < KP; kc += 4) {
        v2f a = *(const v2f*)(arow + kc);
        v2f b = *(const v2f*)(brow + kc);
        // 8 args: (neg_a, A, neg_b, B, c_mod, C, reuse_a, reuse_b)
        acc = __builtin_amdgcn_wmma_f32_16x16x4_f32(
            false, a, false, b, (short)0, acc, false, false);
    }

    // C/D layout: VGPR v -> M = v + 8h, N = lane&15  => t = mbase + v + 8h
    float* p = out + (size_t)(mbase + 8 * h) * N_TOT + (nbase + r);
    if (mbase + 16 <= T_LEN) {
        // Common path (m-tiles 0..36): no guard, 8 stores off one base address;
        // v*N_TOT*4 = v*1MB fits the signed 24-bit immediate offset.
#pragma unroll
        for (int v = 0; v < 8; ++v)
            __builtin_nontemporal_store(acc[v], p + (size_t)v * N_TOT);
    } else {
        // Last m-tile only: guard the T padding rows.
#pragma unroll
        for (int v = 0; v < 8; ++v) {
            int t = mbase + v + 8 * h;
            if (t < T_LEN)
                __builtin_nontemporal_store(acc[v], p + (size_t)v * N_TOT);
        }
    }
}

// ---- launcher --------------------------------------------------------------
extern "C" void kernel_launch(void* const* d_in, const int* in_sizes, int n_in,
                              void* d_out, int out_size, void* d_ws, size_t ws_size,
                              hipStream_t stream) {
    (void)in_sizes; (void)n_in; (void)out_size; (void)ws_size;
    // inputs per reference setup_inputs() order
    // d_in[0] = inp (unused: only fixes seq_len)
    const float* w_v1    = (const float*)d_in[1];
    const int*   rows_v1 = (const int*)  d_in[2];
    const int*   cols_v1 = (const int*)  d_in[3];
    const float* w_lm    = (const float*)d_in[4];
    const int*   rows_lm = (const int*)  d_in[5];
    const int*   cols_lm = (const int*)  d_in[6];
    float* out = (float*)d_out;

    // workspace: [W: N_TOT*KP f32 = 16 MB][S: T_PAD*KP f32]
    float* W = (float*)d_ws;
    float* S = W + (size_t)N_TOT * KP;

    const int n4 = N_TOT * KP / 4;
    zero_w_kernel<<<(n4 + 255) / 256, 256, 0, stream>>>((float4*)W, n4);

    const int nnz = NNZ_V1 + NNZ_LM;
    scatter_w_kernel<<<(nnz + 255) / 256, 256, 0, stream>>>(
        w_v1, rows_v1, cols_v1, w_lm, rows_lm, cols_lm, W);

    gen_spikes_kernel<<<(T_PAD * KP + 255) / 256, 256, 0, stream>>>(S);

    dim3 grid(N_TOT / 16 / 5, T_PAD / 16);   // (3125, 38), exact
    bkg_gemm_kernel<<<grid, 160, 0, stream>>>(S, W, out);
}